// PyQHamiltonianEvolution_1322849927817
// MI455X (gfx1250) — compile-verified
//
#include <hip/hip_runtime.h>

typedef __attribute__((ext_vector_type(2))) float v2f;
typedef __attribute__((ext_vector_type(8))) float v8f;

// D = A(16x4, f32) * B(4x16, f32) + C(16x16, f32) on the CDNA5 matrix pipe.
__device__ __forceinline__ v8f wmma4(v2f a, v2f b, v8f c) {
  return __builtin_amdgcn_wmma_f32_16x16x4_f32(
      /*neg_a=*/false, a, /*neg_b=*/false, b,
      /*c_mod=*/(short)0, c, /*reuse_a=*/false, /*reuse_b=*/false);
}

// Complex-index offset contributed by the 4 support-qubit bits of c/a.
// c bit3 -> qubit 3 (stride 2^20), bit2 -> q7 (2^16), bit1 -> q12 (2^11), bit0 -> q18 (2^5)
__device__ __forceinline__ int c_off(int c) {
  return (((c >> 3) & 1) << 20) + (((c >> 2) & 1) << 16) +
         (((c >> 1) & 1) << 11) + ((c & 1) << 5);
}

// Complex-index offset contributed by the 16 "rest" qubit bits of r.
// r MSB <-> qubit 0 ... r LSB <-> qubit 19 (skipping support qubits).
__device__ __forceinline__ int r_off(int r) {
  int o = 0;
  o += ((r >> 0)  & 1) << 4;   // q19
  o += ((r >> 1)  & 1) << 6;   // q17
  o += ((r >> 2)  & 1) << 7;   // q16
  o += ((r >> 3)  & 1) << 8;   // q15
  o += ((r >> 4)  & 1) << 9;   // q14
  o += ((r >> 5)  & 1) << 10;  // q13
  o += ((r >> 6)  & 1) << 12;  // q11
  o += ((r >> 7)  & 1) << 13;  // q10
  o += ((r >> 8)  & 1) << 14;  // q9
  o += ((r >> 9)  & 1) << 15;  // q8
  o += ((r >> 10) & 1) << 17;  // q6
  o += ((r >> 11) & 1) << 18;  // q5
  o += ((r >> 12) & 1) << 19;  // q4
  o += ((r >> 13) & 1) << 21;  // q2
  o += ((r >> 14) & 1) << 22;  // q1
  o += ((r >> 15) & 1) << 23;  // q0
  return o;
}

// ---------------- Phase 1: U[b] = expm(-i t_b H_b), 16x16 complex ----------------
// One block per batch b; thread (i,j) owns one matrix element.
// Scaling-and-squaring: A = -i t H / 2^8, Taylor order 12, then 8 squarings.
__global__ __launch_bounds__(256) void expm16(const float* __restrict__ h_re,
                                              const float* __restrict__ h_im,
                                              const float* __restrict__ t,
                                              float* __restrict__ Ur,
                                              float* __restrict__ Ui) {
  const int b = blockIdx.x;
  const int tid = threadIdx.x;
  const int i = tid >> 4, j = tid & 15;
  const int ij = i * 17 + j;

  __shared__ float Ar[16 * 17], Ai[16 * 17];
  __shared__ float Tr[16 * 17], Ti[16 * 17];
  __shared__ float Er[16 * 17], Ei[16 * 17];

  const float tf = t[b] * (1.0f / 256.0f);
  const float hr = h_re[(i * 16 + j) * 16 + b];
  const float hi = h_im[(i * 16 + j) * 16 + b];
  // A = -i*(t/256)*(hr + i*hi) = (t/256)*hi - i*(t/256)*hr
  const float ar = tf * hi;
  const float ai = -tf * hr;
  Ar[ij] = ar;  Ai[ij] = ai;
  Tr[ij] = ar;  Ti[ij] = ai;
  Er[ij] = (i == j ? 1.0f : 0.0f) + ar;
  Ei[ij] = ai;
  __syncthreads();

  // Taylor: T <- T*A/k ; E += T
  for (int k = 2; k <= 12; ++k) {
    float pr = 0.0f, pi = 0.0f;
    #pragma unroll
    for (int c = 0; c < 16; ++c) {
      const float xr = Tr[i * 17 + c], xi = Ti[i * 17 + c];
      const float yr = Ar[c * 17 + j], yi = Ai[c * 17 + j];
      pr = fmaf(xr, yr, fmaf(-xi, yi, pr));
      pi = fmaf(xr, yi, fmaf(xi, yr, pi));
    }
    const float inv = 1.0f / (float)k;
    pr *= inv; pi *= inv;
    __syncthreads();
    Tr[ij] = pr;  Ti[ij] = pi;
    Er[ij] += pr; Ei[ij] += pi;
    __syncthreads();
  }

  // 8 squarings: E <- E*E
  for (int s = 0; s < 8; ++s) {
    float pr = 0.0f, pi = 0.0f;
    #pragma unroll
    for (int c = 0; c < 16; ++c) {
      const float xr = Er[i * 17 + c], xi = Ei[i * 17 + c];
      const float yr = Er[c * 17 + j], yi = Ei[c * 17 + j];
      pr = fmaf(xr, yr, fmaf(-xi, yi, pr));
      pi = fmaf(xr, yi, fmaf(xi, yr, pi));
    }
    __syncthreads();
    Er[ij] = pr; Ei[ij] = pi;
    __syncthreads();
  }

  Ur[b * 256 + i * 16 + j] = Er[ij];   // [b][a][c]
  Ui[b * 256 + i * 16 + j] = Ei[ij];
}

// ---------------- Phase 2: out[a,r,b] = sum_c U[b,a,c] * st[c,r,b] ----------------
// Block = 256 threads (8 waves): tile = 16 r-columns x 16 b x full c.
// Each wave runs the complex 16x16 GEMM for two batches via V_WMMA_F32_16X16X4_F32.
#define USTRIDE 272   // sUr/sUi: [b*272 + a*17 + c]
#define SSTRIDE 273   // sSr/sSi: [c*273 + r*17 + b], reused as O[a*273 + r*17 + b]

__global__ __launch_bounds__(256) void apply_u(const float* __restrict__ Ur,
                                               const float* __restrict__ Ui,
                                               const float* __restrict__ s_re,
                                               const float* __restrict__ s_im,
                                               float2* __restrict__ out) {
  __shared__ float sUr[16 * USTRIDE], sUi[16 * USTRIDE];
  __shared__ float sSr[16 * SSTRIDE], sSi[16 * SSTRIDE];

  const int tid = threadIdx.x;

  // Stage all 16 U matrices into LDS (L2-hot, tiny).
  for (int k = tid; k < 4096; k += 256) {
    const int b = k >> 8, rem = k & 255;
    const int a = rem >> 4, c = rem & 15;
    sUr[b * USTRIDE + a * 17 + c] = Ur[k];
    sUi[b * USTRIDE + a * 17 + c] = Ui[k];
  }

  // Coalesced state load: lanes carry (r, b) with b innermost -> 128B lines.
  const int b  = tid & 15;
  const int rl = (tid >> 4) & 15;
  const int rfull = blockIdx.x * 16 + rl;
  const int base = b + r_off(rfull);   // complex index base for this (r, b)
  #pragma unroll
  for (int c = 0; c < 16; ++c) {
    const int addr = base + c_off(c);
    sSr[c * SSTRIDE + rl * 17 + b] = s_re[addr];
    sSi[c * SSTRIDE + rl * 17 + b] = s_im[addr];
  }
  __syncthreads();

  const int lane = tid & 31;
  const int wv   = tid >> 5;          // 0..7 (uniform per wave)
  const int row  = lane & 15;         // A row / B column / D column
  const int hif  = lane >> 4;         // 0: K pair {0,1}; 1: K pair {2,3}

  v8f accR[2], accI[2];
  #pragma unroll
  for (int bb = 0; bb < 2; ++bb) {
    const int mb = wv + 8 * bb;       // batch handled by this wave (uniform)
    v8f aR = {}; v8f aI = {};
    #pragma unroll
    for (int kk = 0; kk < 4; ++kk) {
      const int c0 = kk * 4 + hif * 2;
      const float* urp = &sUr[mb * USTRIDE + row * 17];
      const float* uip = &sUi[mb * USTRIDE + row * 17];
      v2f Afr = {urp[c0], urp[c0 + 1]};
      v2f Afi = {uip[c0], uip[c0 + 1]};
      v2f AfiN = -Afi;                 // wmma4 fixes neg_a=false; negate in VALU
      v2f Bfr = {sSr[c0 * SSTRIDE + row * 17 + mb], sSr[(c0 + 1) * SSTRIDE + row * 17 + mb]};
      v2f Bfi = {sSi[c0 * SSTRIDE + row * 17 + mb], sSi[(c0 + 1) * SSTRIDE + row * 17 + mb]};
      aR = wmma4(Afr,  Bfr, aR);       // + Ur*Sr
      aR = wmma4(AfiN, Bfi, aR);       // - Ui*Si
      aI = wmma4(Afr,  Bfi, aI);       // + Ur*Si
      aI = wmma4(Afi,  Bfr, aI);       // + Ui*Sr
    }
    accR[bb] = aR; accI[bb] = aI;
  }
  __syncthreads();  // all waves done reading sSr/sSi before reuse as output tile

  // Scatter D fragments to LDS as O[a][r][b] (regain b-innermost order).
  #pragma unroll
  for (int bb = 0; bb < 2; ++bb) {
    const int mb = wv + 8 * bb;
    #pragma unroll
    for (int v = 0; v < 8; ++v) {
      const int a = v + 8 * hif;       // D layout: VGPR v, lanes 0-15 -> M=v, 16-31 -> M=v+8
      sSr[a * SSTRIDE + row * 17 + mb] = accR[bb][v];
      sSi[a * SSTRIDE + row * 17 + mb] = accI[bb][v];
    }
  }
  __syncthreads();

  // Coalesced interleaved-complex store: 16 lanes x float2 = 128B contiguous runs.
  #pragma unroll
  for (int a = 0; a < 16; ++a) {
    const int I = base + c_off(a);
    out[I] = make_float2(sSr[a * SSTRIDE + rl * 17 + b],
                         sSi[a * SSTRIDE + rl * 17 + b]);
  }
}

extern "C" void kernel_launch(void* const* d_in, const int* in_sizes, int n_in,
                              void* d_out, int out_size, void* d_ws, size_t ws_size,
                              hipStream_t stream) {
  const float* h_re = (const float*)d_in[0];
  const float* h_im = (const float*)d_in[1];
  const float* t    = (const float*)d_in[2];
  const float* s_re = (const float*)d_in[3];
  const float* s_im = (const float*)d_in[4];

  float* Ur = (float*)d_ws;            // 16*256 floats
  float* Ui = Ur + 16 * 256;           // 16*256 floats (32KB total workspace)

  expm16<<<dim3(16), dim3(256), 0, stream>>>(h_re, h_im, t, Ur, Ui);
  apply_u<<<dim3(4096), dim3(256), 0, stream>>>(Ur, Ui, s_re, s_im, (float2*)d_out);
}